// FlowmatchingActionHead_82592221102552
// MI455X (gfx1250) — compile-verified
//
#include <hip/hip_runtime.h>
#include <hip/hip_bf16.h>
#include <stdint.h>

// Problem dims
#define B_   128
#define T_   16
#define NC_  32
#define SDIM 64
#define ADIM 32
#define D_   1024

// GEMM tiling
#define KC 64   // K chunk staged in LDS
#define NT 32   // columns per block (2 col-tiles of 16)
#define RB 64   // grouped rows per block iteration (4 row-tiles of 16)
#define PA 68   // A LDS pitch (floats): 4m mod 64 spread -> conflict-free reads
#define PB 40   // B LDS pitch (floats): 2*40 mod 64 = 16 -> half-waves disjoint

typedef float v2f __attribute__((ext_vector_type(2)));
typedef float v8f __attribute__((ext_vector_type(8)));
typedef unsigned int v4u __attribute__((ext_vector_type(4)));
typedef int v8i __attribute__((ext_vector_type(8)));
typedef int v4i __attribute__((ext_vector_type(4)));

// ---------------------------------------------------------------------------
// Kernel 0: build per-category batch lists (order-independent result).
// ---------------------------------------------------------------------------
__global__ void build_cats_kernel(const int* __restrict__ cat_ids,
                                  int* __restrict__ cat_count,
                                  int* __restrict__ cat_list) {
  int tid = threadIdx.x;
  if (tid < NC_) cat_count[tid] = 0;
  __syncthreads();
  if (tid < B_) {
    int c = cat_ids[tid];
    int pos = atomicAdd(&cat_count[c], 1);
    cat_list[c * B_ + pos] = tid;
  }
}

// ---------------------------------------------------------------------------
// Kernel 1: per-batch prep: s1 = relu(state*se_W1+b1), tau table,
//           a_emb = actions*ae_W1 + b1. Small-K work; VALU is fine.
// ---------------------------------------------------------------------------
__global__ __launch_bounds__(256)
void prep_kernel(const float* __restrict__ state, const float* __restrict__ actions,
                 const int* __restrict__ timesteps, const int* __restrict__ cat_ids,
                 const float* __restrict__ se_W1, const float* __restrict__ se_b1,
                 const float* __restrict__ ae_W1, const float* __restrict__ ae_b1,
                 float* __restrict__ tau, float* __restrict__ s1,
                 float* __restrict__ a_emb) {
  __shared__ float s_state[SDIM];
  __shared__ float s_act[T_ * ADIM];
  const int b = blockIdx.x;
  const int tid = threadIdx.x;
  const int c = cat_ids[b];

  if (tid < SDIM) s_state[tid] = state[b * SDIM + tid];
  for (int i = tid; i < T_ * ADIM; i += blockDim.x)
    s_act[i] = actions[b * T_ * ADIM + i];
  __syncthreads();

  const float t = (float)timesteps[b];
  const int half = D_ / 2;
  for (int i = tid; i < D_; i += blockDim.x) {
    int j = (i < half) ? i : (i - half);
    float freq = __expf(-(float)j * (9.210340371976184f / (float)half)); // ln(1e4)/half
    float ph = t * freq;
    tau[(size_t)b * D_ + i] = (i < half) ? __sinf(ph) : __cosf(ph);
  }

  for (int n = tid; n < D_; n += blockDim.x) {
    float acc = se_b1[(size_t)c * D_ + n];
    const float* w = se_W1 + (size_t)c * SDIM * D_ + n;
    #pragma unroll 8
    for (int k = 0; k < SDIM; ++k) acc += s_state[k] * w[(size_t)k * D_];
    s1[(size_t)b * D_ + n] = fmaxf(acc, 0.0f);
  }

  for (int tn = tid; tn < T_ * D_; tn += blockDim.x) {
    int tt = tn / D_, n = tn % D_;
    float acc = ae_b1[(size_t)c * D_ + n];
    const float* w = ae_W1 + (size_t)c * ADIM * D_ + n;
    const float* av = &s_act[tt * ADIM];
    #pragma unroll 8
    for (int k = 0; k < ADIM; ++k) acc += av[k] * w[(size_t)k * D_];
    a_emb[((size_t)b * T_ + tt) * D_ + n] = acc;
  }
}

// ---------------------------------------------------------------------------
// Grouped-GEMM A-element gather (fuses concat / pos_emb add). MODE/R static.
// ---------------------------------------------------------------------------
template <int MODE, int R>
__device__ __forceinline__ float fetchA(int c, int gl, int cnt,
                                        const int* __restrict__ cat_list, int k,
                                        const float* __restrict__ s1,
                                        const float* __restrict__ a_emb,
                                        const float* __restrict__ tau,
                                        const float* __restrict__ x2s,
                                        const float* __restrict__ s2,
                                        const float* __restrict__ x3,
                                        const float* __restrict__ pos_emb,
                                        const float* __restrict__ h) {
  int bi = gl / R;
  if (bi >= cnt) return 0.0f;
  int r = gl - bi * R;
  int b = cat_list[c * B_ + bi];
  if (MODE == 0) return s1[(size_t)b * D_ + k];
  if (MODE == 1) return (k < D_) ? a_emb[((size_t)b * T_ + r) * D_ + k]
                                 : tau[(size_t)b * D_ + (k - D_)];
  if (MODE == 2) return x2s[((size_t)b * T_ + r) * D_ + k];
  if (MODE == 3) {
    float v = (r == 0) ? s2[(size_t)b * D_ + k]
                       : x3[((size_t)b * T_ + (r - 1)) * D_ + k];
    return v + pos_emb[(size_t)r * D_ + k];
  }
  return h[((size_t)b * (T_ + 1) + r) * D_ + k];
}

// ---------------------------------------------------------------------------
// Kernel 2: grouped GEMM, WMMA f32 16x16x4.
// Grid: (Ncols/NT, NC). Block: 256 threads = 8 waves = 4 row-tiles x 2 col-tiles.
// Weight chunks are pulled LDS-direct by the Tensor Data Mover (one 2D D# per
// wave covering 8 rows), with TDM hardware padding producing the pitch-40,
// bank-conflict-free layout. A chunks are gathered via VALU into pitch-68 LDS.
// ---------------------------------------------------------------------------
template <int MODE, int R>
__global__ __launch_bounds__(256)
void grouped_gemm_wmma(int K, int Ncols,
                       const float* __restrict__ W, const float* __restrict__ bias,
                       const int* __restrict__ cat_count,
                       const int* __restrict__ cat_list,
                       const float* __restrict__ s1, const float* __restrict__ a_emb,
                       const float* __restrict__ tau, const float* __restrict__ x2s,
                       const float* __restrict__ s2, const float* __restrict__ x3,
                       const float* __restrict__ pos_emb, const float* __restrict__ h,
                       float* __restrict__ dst) {
  __shared__ __align__(16) float As[RB * PA];  // ~17 KB
  __shared__ __align__(16) float Bs[KC * PB];  // ~10 KB

  const int c = blockIdx.y;
  const int cnt = cat_count[c];
  if (cnt == 0) return;
  const int rows = cnt * R;
  const int n0 = blockIdx.x * NT;

  const int tid = threadIdx.x;
  const int lane = tid & 31;
  const int wave = tid >> 5;
  const int row_tile = wave >> 1;         // 0..3
  const int col_tile = wave & 1;          // 0..1
  const int m_lane = lane & 15;           // M (A) or N (B/C) index within tile
  const int koff = (lane >> 4) * 2;       // K offset per half-wave (A/B layout)

  const size_t wbase = (size_t)c * (size_t)K * (size_t)Ncols;

  for (int rb = 0; rb < rows; rb += RB) {
    v8f acc = {0.f, 0.f, 0.f, 0.f, 0.f, 0.f, 0.f, 0.f};

    for (int kc = 0; kc < K; kc += KC) {
      __syncthreads();

#if __has_builtin(__builtin_amdgcn_tensor_load_to_lds)
      // --- TDM: each wave DMA-loads 8 weight rows (32 dwords each) into LDS,
      //     hardware-padded to pitch 40 dwords (pad_interval=32dw, amount=8dw).
      {
        const float* gsrc = W + wbase + (size_t)(kc + wave * 8) * Ncols + n0;
        uint64_t ga = (uint64_t)(uintptr_t)gsrc;
        uint32_t laddr = (uint32_t)(uintptr_t)(&Bs[wave * 8 * PB]);
        v4u g0 = {1u,                                        // count=1, user mode
                  laddr,                                     // lds_addr
                  (uint32_t)ga,                              // global_addr[31:0]
                  (((uint32_t)(ga >> 32)) & 0x01FFFFFFu) | (2u << 30)}; // +type=2
        v8i g1;
        g1[0] = (int)((2u << 16)      // data_size = 4B
                    | (1u << 20)      // pad_enable
                    | (4u << 22)      // pad_interval: 32 dwords
                    | (7u << 25));    // pad_amount:   8 dwords
        g1[1] = (int)(32u << 16);     // tensor_dim0 = 32 (low16 in [63:48])
        g1[2] = (int)(8u << 16);      // tensor_dim1 = 8
        g1[3] = (int)(32u << 16);     // tile_dim0 = 32
        g1[4] = 8;                    // tile_dim1 = 8, tile_dim2 = 0
        g1[5] = Ncols;                // tensor_dim0_stride (data_size units)
        g1[6] = 0;
        g1[7] = 0;
        v4i z4 = {0, 0, 0, 0};
        v8i z8 = {0, 0, 0, 0, 0, 0, 0, 0};
        __builtin_amdgcn_tensor_load_to_lds(g0, g1, z4, z4, z8, 0);
      }
#else
      for (int i = tid; i < KC * NT; i += 256) {
        int kk = i / NT, nn = i % NT;
        Bs[kk * PB + nn] = W[wbase + (size_t)(kc + kk) * Ncols + (n0 + nn)];
      }
#endif

      // --- Stage grouped-A chunk (gather fuses concat / pos_emb), pitch 68.
      for (int i = tid; i < RB * KC; i += 256) {
        int rr = i / KC, kk = i % KC;
        As[rr * PA + kk] = fetchA<MODE, R>(c, rb + rr, cnt, cat_list, kc + kk,
                                           s1, a_emb, tau, x2s, s2, x3, pos_emb, h);
      }

#if __has_builtin(__builtin_amdgcn_tensor_load_to_lds)
      __builtin_amdgcn_s_wait_tensorcnt(0);
#endif
      __syncthreads();

      const float* Arow = &As[(row_tile * 16 + m_lane) * PA];
      const float* Bcol = &Bs[col_tile * 16 + m_lane];
      #pragma unroll
      for (int ks = 0; ks < KC; ks += 4) {
        v2f a, bf;
        a[0]  = Arow[ks + koff];
        a[1]  = Arow[ks + koff + 1];
        bf[0] = Bcol[(ks + koff) * PB];
        bf[1] = Bcol[(ks + koff + 1) * PB];
        acc = __builtin_amdgcn_wmma_f32_16x16x4_f32(
            /*neg_a=*/false, a, /*neg_b=*/false, bf,
            /*c_mod=*/(short)0, acc, /*reuse_a=*/false, /*reuse_b=*/false);
      }
    }

    // Epilogue: bias + activation + scatter back to per-batch layout
    #pragma unroll
    for (int v = 0; v < 8; ++v) {
      int m = v + 8 * (lane >> 4);
      int gl = rb + row_tile * 16 + m;
      int col = n0 + col_tile * 16 + m_lane;
      if (gl < rows && col < Ncols) {
        int bi = gl / R;
        int r = gl - bi * R;
        int b = cat_list[c * B_ + bi];
        float val = acc[v] + bias[(size_t)c * Ncols + col];
        if (MODE == 1) {                       // swish
          float sg = 1.0f / (1.0f + __expf(-val));
          val *= sg;
        } else if (MODE == 3) {                // relu
          val = fmaxf(val, 0.0f);
        }
        size_t idx;
        if (MODE == 0)                    idx = (size_t)b * D_ + col;
        else if (MODE == 1 || MODE == 2)  idx = ((size_t)b * T_ + r) * D_ + col;
        else if (MODE == 3)               idx = ((size_t)b * (T_ + 1) + r) * D_ + col;
        else                              idx = ((size_t)b * (T_ + 1) + r) * ADIM + col;
        dst[idx] = val;
      }
    }
  }
}

// ---------------------------------------------------------------------------
// Host-side launch
// ---------------------------------------------------------------------------
extern "C" void kernel_launch(void* const* d_in, const int* in_sizes, int n_in,
                              void* d_out, int out_size, void* d_ws, size_t ws_size,
                              hipStream_t stream) {
  (void)in_sizes; (void)n_in; (void)out_size; (void)ws_size;
  const float* state     = (const float*)d_in[0];
  const float* actions   = (const float*)d_in[1];
  const int*   timesteps = (const int*)d_in[2];
  const int*   cat_ids   = (const int*)d_in[3];
  const float* pos_emb   = (const float*)d_in[4];
  const float* se_W1     = (const float*)d_in[5];
  const float* se_b1     = (const float*)d_in[6];
  const float* se_W2     = (const float*)d_in[7];
  const float* se_b2     = (const float*)d_in[8];
  const float* ae_W1     = (const float*)d_in[9];
  const float* ae_b1     = (const float*)d_in[10];
  const float* ae_W2     = (const float*)d_in[11];
  const float* ae_b2     = (const float*)d_in[12];
  const float* ae_W3     = (const float*)d_in[13];
  const float* ae_b3     = (const float*)d_in[14];
  const float* ad_W1     = (const float*)d_in[15];
  const float* ad_b1     = (const float*)d_in[16];
  const float* ad_W2     = (const float*)d_in[17];
  const float* ad_b2     = (const float*)d_in[18];
  float* out = (float*)d_out;

  // Workspace layout
  char* ws = (char*)d_ws;
  int* cat_count = (int*)ws;  ws += 256;                 // 32 ints (padded)
  int* cat_list  = (int*)ws;  ws += NC_ * B_ * sizeof(int);
  ws = (char*)(((uintptr_t)ws + 255) & ~(uintptr_t)255);
  float* tau = (float*)ws;  ws += (size_t)B_ * D_ * 4;
  float* s1  = (float*)ws;  ws += (size_t)B_ * D_ * 4;
  float* aem = (float*)ws;  ws += (size_t)B_ * T_ * D_ * 4;
  float* s2  = (float*)ws;  ws += (size_t)B_ * D_ * 4;
  float* x2s = (float*)ws;  ws += (size_t)B_ * T_ * D_ * 4;
  float* x3  = (float*)ws;  ws += (size_t)B_ * T_ * D_ * 4;
  float* h   = (float*)ws;  ws += (size_t)B_ * (T_ + 1) * D_ * 4;

  build_cats_kernel<<<1, 128, 0, stream>>>(cat_ids, cat_count, cat_list);
  prep_kernel<<<B_, 256, 0, stream>>>(state, actions, timesteps, cat_ids,
                                      se_W1, se_b1, ae_W1, ae_b1, tau, s1, aem);

  const dim3 blk(256);
  // mode 0: s2 = s1 @ se_W2 + se_b2                (M=cnt,   K=1024, N=1024)
  grouped_gemm_wmma<0, 1><<<dim3(D_ / NT, NC_), blk, 0, stream>>>(
      D_, D_, se_W2, se_b2, cat_count, cat_list,
      s1, aem, tau, x2s, s2, x3, pos_emb, h, s2);
  // mode 1: x2 = swish([a_emb|tau] @ ae_W2 + b2)   (M=16cnt, K=2048, N=1024)
  grouped_gemm_wmma<1, T_><<<dim3(D_ / NT, NC_), blk, 0, stream>>>(
      2 * D_, D_, ae_W2, ae_b2, cat_count, cat_list,
      s1, aem, tau, x2s, s2, x3, pos_emb, h, x2s);
  // mode 2: x3 = x2 @ ae_W3 + b3                   (M=16cnt, K=1024, N=1024)
  grouped_gemm_wmma<2, T_><<<dim3(D_ / NT, NC_), blk, 0, stream>>>(
      D_, D_, ae_W3, ae_b3, cat_count, cat_list,
      s1, aem, tau, x2s, s2, x3, pos_emb, h, x3);
  // mode 3: h = relu((concat(s2,x3)+pos) @ ad_W1)  (M=17cnt, K=1024, N=1024)
  grouped_gemm_wmma<3, T_ + 1><<<dim3(D_ / NT, NC_), blk, 0, stream>>>(
      D_, D_, ad_W1, ad_b1, cat_count, cat_list,
      s1, aem, tau, x2s, s2, x3, pos_emb, h, h);
  // mode 4: out = h @ ad_W2 + b2                   (M=17cnt, K=1024, N=32)
  grouped_gemm_wmma<4, T_ + 1><<<dim3(1, NC_), blk, 0, stream>>>(
      D_, ADIM, ad_W2, ad_b2, cat_count, cat_list,
      s1, aem, tau, x2s, s2, x3, pos_emb, h, out);
}